// HungarianLoss_592705487245
// MI455X (gfx1250) — compile-verified
//
#include <hip/hip_runtime.h>
#include <math.h>

#define B_ 128
#define Q_ 900
#define M_ 100
#define C_ 7
#define T_ 256

typedef __attribute__((ext_vector_type(2))) float v2f;
typedef __attribute__((ext_vector_type(8))) float v8f;

// ---------------------------------------------------------------------------
// Kernel 1: cost matrix via WMMA.
// class_cost = (-logP)(Q x C) @ onehot(C x M) done as V_WMMA_F32_16X16X4_F32
// (two K-steps: K=0..3 and K=4..7 with K=7 zero-padded).  Box L1 (first 3
// dims) added in VALU from registers hoisted out of the m-tile loop.
// Output stored TRANSPOSED: costT[b][m][q] so Hungarian row scans are
// contiguous/coalesced; per-lane D values are 8 contiguous floats -> two
// b128 stores on the full-tile fast path.
// One wave (32 threads) computes a 16(q) x 112(m) strip.
// ---------------------------------------------------------------------------
__global__ __launch_bounds__(32)
void cost_kernel(const float* __restrict__ logits,
                 const float* __restrict__ pboxes,
                 const int*   __restrict__ glabels,
                 const float* __restrict__ gboxes,
                 float* __restrict__ costT)
{
    const int lane  = threadIdx.x;      // 0..31
    const int half  = lane >> 4;        // 0: K 0/1 (and 4/5), 1: K 2/3 (and 6/7)
    const int r     = lane & 15;
    const int b     = blockIdx.y;
    const int qbase = blockIdx.x * 16;

    // ---- A fragment: -log_softmax row q = qbase + r (clamped; tail masked) --
    int qa = qbase + r; if (qa >= Q_) qa = Q_ - 1;
    const float* lp = logits + ((size_t)b * Q_ + qa) * C_;
    float l[7];
    #pragma unroll
    for (int c = 0; c < C_; ++c) l[c] = lp[c];
    float mx = l[0];
    #pragma unroll
    for (int c = 1; c < C_; ++c) mx = fmaxf(mx, l[c]);
    float se = 0.f;
    #pragma unroll
    for (int c = 0; c < C_; ++c) se += expf(l[c] - mx);
    const float lse = mx + logf(se);
    float a[8];
    #pragma unroll
    for (int c = 0; c < C_; ++c) a[c] = lse - l[c];   // -log p
    a[7] = 0.f;                                       // K=7 pad

    // 16x4 f32 A layout: lanes 0-15 hold K={0,1}, lanes 16-31 hold K={2,3}
    v2f A0, A1;
    A0.x = half ? a[2] : a[0];
    A0.y = half ? a[3] : a[1];
    A1.x = half ? a[6] : a[4];
    A1.y = half ? a[7] : a[5];

    // ---- hoist pred-box xyz for this lane's 8 D rows (invariant over m) ----
    float pbx[8], pby[8], pbz[8];
    #pragma unroll
    for (int v = 0; v < 8; ++v) {
        const int q2 = qbase + v + half * 8;
        const int qc = (q2 < Q_) ? q2 : (Q_ - 1);
        const float* pb = pboxes + ((size_t)b * Q_ + qc) * 7;
        pbx[v] = pb[0]; pby[v] = pb[1]; pbz[v] = pb[2];
    }
    const bool fullq = (qbase + 16 <= Q_);   // 56 of 57 q-tiles

    // ---- loop over 7 m-tiles of 16 (112 >= M=100, tail masked) -------------
    for (int mt = 0; mt < 7; ++mt) {
        const int  mm = mt * 16 + r;          // this lane's N column
        const bool vm = (mm < M_);
        const int  mi = vm ? mm : (M_ - 1);
        int lab = glabels[b * M_ + mi];
        if (!vm) lab = -1;

        // B one-hot fragment, 4x16: lanes 0-15 K={0,1}, lanes 16-31 K={2,3}
        const int k0 = half * 2;
        v2f B0, B1;
        B0.x = (lab == k0)     ? 1.f : 0.f;
        B0.y = (lab == k0 + 1) ? 1.f : 0.f;
        B1.x = (lab == k0 + 4) ? 1.f : 0.f;
        B1.y = (lab == k0 + 5) ? 1.f : 0.f;   // k0+5==7 never matches (lab<7)

        v8f acc = {};
        acc = __builtin_amdgcn_wmma_f32_16x16x4_f32(
                  false, A0, false, B0, (short)0, acc, false, false);
        acc = __builtin_amdgcn_wmma_f32_16x16x4_f32(
                  false, A1, false, B1, (short)0, acc, false, false);

        const float* gp = gboxes + ((size_t)b * M_ + mi) * 7;
        const float gb0 = gp[0], gb1 = gp[1], gb2 = gp[2];

        // D layout: VGPR v -> row M = v + half*8, N = lane&15
        float outv[8];
        #pragma unroll
        for (int v = 0; v < 8; ++v)
            outv[v] = acc[v] + fabsf(pbx[v] - gb0) + fabsf(pby[v] - gb1)
                             + fabsf(pbz[v] - gb2);

        float* dst = costT + ((size_t)b * M_ + mi) * Q_ + qbase + half * 8;
        if (vm && fullq) {
            // contiguous, 16B-aligned: two 128-bit stores
            ((float4*)dst)[0] = make_float4(outv[0], outv[1], outv[2], outv[3]);
            ((float4*)dst)[1] = make_float4(outv[4], outv[5], outv[6], outv[7]);
        } else {
            #pragma unroll
            for (int v = 0; v < 8; ++v) {
                const int q2 = qbase + v + half * 8;
                if (vm && q2 < Q_) dst[v] = outv[v];
            }
        }
    }
}

// ---------------------------------------------------------------------------
// Kernel 2: exact Jonker-Volgenant assignment, one workgroup per batch.
// Cost is transposed (n=M=100 rows <= m=Q=900 cols), matching scipy.
// Dual variables in double (matches reference float64); fp32 cost promoted
// exactly.  Argmin uses lowest-index tie-break (np.argmin semantics).
// ---------------------------------------------------------------------------
__global__ __launch_bounds__(T_)
void hungarian_kernel(const float* __restrict__ costT, int* __restrict__ match)
{
    const int b   = blockIdx.x;
    const int tid = threadIdx.x;
    const int n   = M_;
    const int m   = Q_;

    __shared__ double u[M_ + 1];
    __shared__ double vv[Q_ + 1];
    __shared__ double minv[Q_ + 1];
    __shared__ int    way[Q_ + 1];
    __shared__ int    used[Q_ + 1];
    __shared__ int    p[Q_ + 1];
    __shared__ double rv[T_];
    __shared__ int    ri[T_];
    __shared__ int    s_j0, s_i0, s_brk;

    const double INF = __builtin_inf();

    for (int j = tid; j <= m; j += T_) { vv[j] = 0.0; p[j] = 0; }
    for (int i = tid; i <= n; i += T_) u[i] = 0.0;
    __syncthreads();

    for (int i = 1; i <= n; ++i) {
        for (int j = tid; j <= m; j += T_) { minv[j] = INF; used[j] = 0; }
        if (tid == 0) { p[0] = i; s_j0 = 0; s_brk = 0; }
        __syncthreads();

        for (int step = 0; step <= m; ++step) {
            if (tid == 0) {
                used[s_j0] = 1;
                s_i0 = p[s_j0];
            }
            __syncthreads();
            const int    i0  = s_i0;
            const int    j0  = s_j0;
            const double ui0 = u[i0];
            const float* row = costT + ((size_t)b * M_ + (i0 - 1)) * Q_;

            // parallel scan: relax minv over free cols + local argmin
            double bv = INF; int bj = m + 1;
            for (int j = 1 + tid; j <= m; j += T_) {
                if (!used[j]) {
                    const double cur = (double)row[j - 1] - ui0 - vv[j];
                    if (cur < minv[j]) { minv[j] = cur; way[j] = j0; }
                    const double mv = minv[j];
                    if (mv < bv) { bv = mv; bj = j; }
                }
            }
            rv[tid] = bv; ri[tid] = bj;
            __syncthreads();
            for (int s = T_ / 2; s > 0; s >>= 1) {
                if (tid < s) {
                    const double ov = rv[tid + s]; const int oj = ri[tid + s];
                    if (ov < rv[tid] || (ov == rv[tid] && oj < ri[tid])) {
                        rv[tid] = ov; ri[tid] = oj;
                    }
                }
                __syncthreads();
            }
            const double delta = rv[0];
            const int    j0n   = ri[0];

            // dual update (matched rows of used cols are distinct: race-free)
            for (int j = tid; j <= m; j += T_) {
                if (used[j]) { u[p[j]] += delta; vv[j] -= delta; }
                else         { minv[j] -= delta; }
            }
            if (tid == 0) {
                s_j0 = j0n;
                if (p[j0n] == 0) s_brk = 1;
            }
            __syncthreads();
            if (s_brk) break;
        }

        // augment along alternating path (serial, short)
        if (tid == 0) {
            int j0 = s_j0;
            while (j0 != 0) {
                const int j1 = way[j0];
                p[j0] = p[j1];
                j0 = j1;
            }
        }
        __syncthreads();
    }

    // match[b][gt] = query index
    for (int j = 1 + tid; j <= m; j += T_) {
        const int pi = p[j];
        if (pi > 0) match[b * M_ + (pi - 1)] = j - 1;
    }
}

// ---------------------------------------------------------------------------
// Kernel 3: matched CE + L1 loss, deterministic two-stage reduction.
// ---------------------------------------------------------------------------
__global__ __launch_bounds__(256)
void loss_partial_kernel(const float* __restrict__ logits,
                         const float* __restrict__ pboxes,
                         const int*   __restrict__ glabels,
                         const float* __restrict__ gboxes,
                         const int*   __restrict__ match,
                         float* __restrict__ partial)
{
    __shared__ float red[256];
    const int tid = threadIdx.x;
    const int idx = blockIdx.x * 256 + tid;     // b*M + g
    float val = 0.f;
    if (idx < B_ * M_) {
        const int b   = idx / M_;
        const int q   = match[idx];
        const int lab = glabels[idx];
        const float* lpq = logits + ((size_t)b * Q_ + q) * C_;
        float mx = lpq[0];
        #pragma unroll
        for (int c = 1; c < C_; ++c) mx = fmaxf(mx, lpq[c]);
        float se = 0.f;
        #pragma unroll
        for (int c = 0; c < C_; ++c) se += expf(lpq[c] - mx);
        const float ce = (mx + logf(se)) - lpq[lab];

        const float* pb = pboxes + ((size_t)b * Q_ + q) * 7;
        const float* gb = gboxes + (size_t)idx * 7;
        float l1 = 0.f;
        #pragma unroll
        for (int d = 0; d < 7; ++d) l1 += fabsf(pb[d] - gb[d]);

        val = ce * (1.0f / (B_ * M_)) + l1 * (1.0f / (B_ * M_ * 7));
    }
    red[tid] = val;
    __syncthreads();
    for (int s = 128; s > 0; s >>= 1) {
        if (tid < s) red[tid] += red[tid + s];
        __syncthreads();
    }
    if (tid == 0) partial[blockIdx.x] = red[0];
}

__global__ void loss_final_kernel(const float* __restrict__ partial,
                                  int nparts, float* __restrict__ out)
{
    if (threadIdx.x == 0 && blockIdx.x == 0) {
        float s = 0.f;
        for (int i = 0; i < nparts; ++i) s += partial[i];
        out[0] = s;
    }
}

// ---------------------------------------------------------------------------
extern "C" void kernel_launch(void* const* d_in, const int* in_sizes, int n_in,
                              void* d_out, int out_size, void* d_ws, size_t ws_size,
                              hipStream_t stream)
{
    const float* logits  = (const float*)d_in[0];   // (B,Q,7) f32
    const float* pboxes  = (const float*)d_in[1];   // (B,Q,7) f32
    const int*   glabels = (const int*)  d_in[2];   // (B,M)   i32
    const float* gboxes  = (const float*)d_in[3];   // (B,M,7) f32
    float* out = (float*)d_out;

    char* ws = (char*)d_ws;
    float* costT   = (float*)ws;                                      // B*M*Q f32
    size_t off     = (size_t)B_ * M_ * Q_ * sizeof(float);
    int*   match   = (int*)(ws + off);                                // B*M i32
    off           += (size_t)B_ * M_ * sizeof(int);
    float* partial = (float*)(ws + off);                              // <=64 f32

    dim3 gc((Q_ + 15) / 16, B_);
    cost_kernel<<<gc, 32, 0, stream>>>(logits, pboxes, glabels, gboxes, costT);

    hungarian_kernel<<<B_, T_, 0, stream>>>(costT, match);

    const int nitems = B_ * M_;
    const int nblk   = (nitems + 255) / 256;
    loss_partial_kernel<<<nblk, 256, 0, stream>>>(logits, pboxes, glabels,
                                                  gboxes, match, partial);
    loss_final_kernel<<<1, 32, 0, stream>>>(partial, nblk, out);
}